// GCN_31104153158272
// MI455X (gfx1250) — compile-verified
//
#include <hip/hip_runtime.h>

#define N_NODES 100000
#define N_EDGES 3200000
#define F_IN    100
#define HID     16
#define N_CLASS 18

typedef __attribute__((ext_vector_type(2))) float v2f;
typedef __attribute__((ext_vector_type(8))) float v8f;

__device__ __forceinline__ v8f wmma4(v2f a, v2f b, v8f c) {
  // V_WMMA_F32_16X16X4_F32: D(16x16 f32) = A(16x4 f32) * B(4x16 f32) + C
  // 8-arg pattern: (neg_a, A, neg_b, B, c_mod, C, reuse_a, reuse_b)
  return __builtin_amdgcn_wmma_f32_16x16x4_f32(false, a, false, b, (short)0, c,
                                               false, false);
}

// out[M x 16] = in[M x K] @ W[K x 16] (+ bias) (relu). One wave per 16-row tile.
// Grid must cover exactly M/16 tiles (M % 16 == 0) so EXEC is all-1s at WMMA.
template <int K, bool BIAS, bool RELU>
__global__ void gemm16_wmma(const float* __restrict__ in,
                            const float* __restrict__ W,
                            const float* __restrict__ bias,
                            float* __restrict__ out) {
  const int lane = threadIdx.x & 31;
  const int wid  = threadIdx.x >> 5;
  const int tile = blockIdx.x * (blockDim.x >> 5) + wid;
  const int half = lane >> 4;   // 0: K pair (k,k+1) / M rows 0..7 of D
  const int l16  = lane & 15;

  const float* xr = in + (size_t)(tile * 16 + l16) * K;  // A: row per lane
  v8f c = {0.f, 0.f, 0.f, 0.f, 0.f, 0.f, 0.f, 0.f};

#pragma unroll
  for (int k = 0; k < K; k += 4) {
    const int kk = k + half * 2;
    v2f a; a.x = xr[kk];                 a.y = xr[kk + 1];
    v2f b; b.x = W[kk * HID + l16];      b.y = W[(kk + 1) * HID + l16];
    c = wmma4(a, b, c);
  }

  const float bv = BIAS ? bias[l16] : 0.f;
  const int mbase = tile * 16 + half * 8;     // C/D: VGPR r -> M = mbase + r
  float* op = out + (size_t)mbase * HID + l16;
#pragma unroll
  for (int r = 0; r < 8; ++r) {
    float v = c[r] + bv;
    if (RELU) v = fmaxf(v, 0.f);
    op[(size_t)r * HID] = v;
  }
}

__global__ void deg_init(float* __restrict__ deg) {
  int i = blockIdx.x * blockDim.x + threadIdx.x;
  if (i < N_NODES) deg[i] = 1.0f;  // self-loop
}

__global__ void deg_accum(const int* __restrict__ ei, float* __restrict__ deg) {
  int e = blockIdx.x * blockDim.x + threadIdx.x;
  if (e < N_EDGES) atomicAdd(&deg[ei[N_EDGES + e]], 1.0f);  // dst row
}

__global__ void deg_finish(float* __restrict__ deg) {  // deg -> dinv in place
  int i = blockIdx.x * blockDim.x + threadIdx.x;
  if (i < N_NODES) deg[i] = 1.0f / sqrtf(deg[i]);
}

// agg[n][j] = bias[j] + dinv[n]^2 * t[n][j]   (bias + self-loop message)
__global__ void conv_init(const float* __restrict__ dinv,
                          const float* __restrict__ t,
                          const float* __restrict__ bias,
                          float* __restrict__ agg) {
  int i = blockIdx.x * blockDim.x + threadIdx.x;
  if (i >= N_NODES * HID) return;
  int n = i >> 4, j = i & 15;
  float di = dinv[n];
  agg[i] = bias[j] + di * di * t[i];
}

// agg[dst] += dinv[src]*dinv[dst] * t[src]  (16 floats per edge; L2-resident)
__global__ void conv_scatter(const int* __restrict__ ei,
                             const float* __restrict__ dinv,
                             const float* __restrict__ t,
                             float* __restrict__ agg) {
  int e = blockIdx.x * blockDim.x + threadIdx.x;
  if (e >= N_EDGES) return;
  int s = ei[e];
  int d = ei[N_EDGES + e];
  float w = dinv[s] * dinv[d];
  const float4* ts = (const float4*)(t + (size_t)s * HID);
  float* ad = agg + (size_t)d * HID;
#pragma unroll
  for (int q = 0; q < 4; ++q) {
    float4 v = ts[q];
    atomicAdd(ad + 4 * q + 0, w * v.x);
    atomicAdd(ad + 4 * q + 1, w * v.y);
    atomicAdd(ad + 4 * q + 2, w * v.z);
    atomicAdd(ad + 4 * q + 3, w * v.w);
  }
}

__global__ void relu_ip(float* __restrict__ a, int n) {
  int i = blockIdx.x * blockDim.x + threadIdx.x;
  if (i < n) a[i] = fmaxf(a[i], 0.f);
}

// out[n] = log_softmax(h[n] @ W2 + b2) ; W2: [16 x 18]
__global__ void lin2_lsm(const float* __restrict__ h,
                         const float* __restrict__ W2,
                         const float* __restrict__ b2,
                         float* __restrict__ out) {
  __shared__ float sW[HID * N_CLASS];
  __shared__ float sB[N_CLASS];
  for (int i = threadIdx.x; i < HID * N_CLASS; i += blockDim.x) sW[i] = W2[i];
  if (threadIdx.x < N_CLASS) sB[threadIdx.x] = b2[threadIdx.x];
  __syncthreads();

  int n = blockIdx.x * blockDim.x + threadIdx.x;
  if (n >= N_NODES) return;

  float hv[HID];
  const float4* hp = (const float4*)(h + (size_t)n * HID);
#pragma unroll
  for (int q = 0; q < 4; ++q) {
    float4 v = hp[q];
    hv[4 * q + 0] = v.x; hv[4 * q + 1] = v.y;
    hv[4 * q + 2] = v.z; hv[4 * q + 3] = v.w;
  }

  float logits[N_CLASS];
  float m = -3.402823466e38f;
#pragma unroll
  for (int c = 0; c < N_CLASS; ++c) {
    float acc = sB[c];
#pragma unroll
    for (int k = 0; k < HID; ++k) acc = fmaf(hv[k], sW[k * N_CLASS + c], acc);
    logits[c] = acc;
    m = fmaxf(m, acc);
  }
  float sum = 0.f;
#pragma unroll
  for (int c = 0; c < N_CLASS; ++c) sum += expf(logits[c] - m);
  float lse = m + logf(sum);
  float* op = out + (size_t)n * N_CLASS;
#pragma unroll
  for (int c = 0; c < N_CLASS; ++c) op[c] = logits[c] - lse;
}

extern "C" void kernel_launch(void* const* d_in, const int* in_sizes, int n_in,
                              void* d_out, int out_size, void* d_ws, size_t ws_size,
                              hipStream_t stream) {
  const float* x   = (const float*)d_in[0];
  const int*   ei  = (const int*)d_in[1];   // [2, N_EDGES]: row0=src, row1=dst
  const float* W1  = (const float*)d_in[2];
  const float* b1  = (const float*)d_in[3];
  const float* Wc0 = (const float*)d_in[4];
  const float* bc0 = (const float*)d_in[5];
  const float* Wc1 = (const float*)d_in[6];
  const float* bc1 = (const float*)d_in[7];
  const float* W2  = (const float*)d_in[8];
  const float* b2  = (const float*)d_in[9];
  float* out = (float*)d_out;

  float* A    = (float*)d_ws;                     // N_NODES*16 floats
  float* B    = A + (size_t)N_NODES * HID;        // N_NODES*16 floats
  float* C    = B + (size_t)N_NODES * HID;        // N_NODES*16 floats
  float* dinv = C + (size_t)N_NODES * HID;        // N_NODES floats

  const int T = 256;
  const int gN  = (N_NODES + T - 1) / T;
  const int gE  = (N_EDGES + T - 1) / T;
  const int gF  = (N_NODES * HID + T - 1) / T;
  const int TILES = N_NODES / 16;                 // 6250, exact
  const int gG  = TILES / 2;                      // 2 waves (64 thr) per block

  // degrees -> dinv
  deg_init  <<<gN, T, 0, stream>>>(dinv);
  deg_accum <<<gE, T, 0, stream>>>(ei, dinv);
  deg_finish<<<gN, T, 0, stream>>>(dinv);

  // h = relu(x @ W1 + b1)
  gemm16_wmma<F_IN, true, true><<<gG, 64, 0, stream>>>(x, W1, b1, A);

  // conv1: t = h @ Wc0 ; agg = bc0 + selfloop + scatter ; relu
  gemm16_wmma<HID, false, false><<<gG, 64, 0, stream>>>(A, Wc0, nullptr, B);
  conv_init   <<<gF, T, 0, stream>>>(dinv, B, bc0, C);
  conv_scatter<<<gE, T, 0, stream>>>(ei, dinv, B, C);
  relu_ip     <<<gF, T, 0, stream>>>(C, N_NODES * HID);

  // conv2
  gemm16_wmma<HID, false, false><<<gG, 64, 0, stream>>>(C, Wc1, nullptr, A);
  conv_init   <<<gF, T, 0, stream>>>(dinv, A, bc1, B);
  conv_scatter<<<gE, T, 0, stream>>>(ei, dinv, A, B);
  relu_ip     <<<gF, T, 0, stream>>>(B, N_NODES * HID);

  // lin2 + log_softmax
  lin2_lsm<<<gN, T, 0, stream>>>(B, W2, b2, out);
}